// MutualInformation_48954037239885
// MI455X (gfx1250) — compile-verified
//
#include <hip/hip_runtime.h>
#include <hip/hip_bf16.h>
#include <math.h>

// Mutual-information histogram loss for MI455X (gfx1250, wave32).
// Bandwidth-bound: 128 MB streamed once -> ~5.5 us at 23.3 TB/s HBM.
// Strategy: non-temporal b128 streaming loads + per-8-lane LDS sub-histograms
// (ds_add_u32, bank-padded stride 65) + global_atomic_add_u32 merge + finalize.

#define NB        64         // bins (matches reference NUM_BINS)
#define NBATCH    16
#define NSUB      32         // one sub-histogram per 8 lanes of a 256-thread block
#define SUBSTRIDE 65         // 64 bins + 1 pad dword: bank = (s + bin) & 63
#define MI_EPS    1e-8f

// Native clang vector type: __builtin_nontemporal_load requires a scalar or
// vector-of-scalar type, not HIP's HIP_vector_type<float,4> wrapper class.
typedef float v4f __attribute__((ext_vector_type(4)));

__global__ void mi_hist_zero_kernel(unsigned int* __restrict__ gh) {
    int i = blockIdx.x * blockDim.x + threadIdx.x;
    if (i < NBATCH * NB) gh[i] = 0u;
}

__device__ __forceinline__ void mi_bin_one(float xv, float yv,
                                           unsigned int* __restrict__ sub) {
    // v = x*nb + y ; histc over [0, 64]: drop outside, v==64 -> last bin.
    float v = fmaf(xv, 64.0f, yv);
    if (v >= 0.0f && v <= 64.0f) {
        int idx = (int)v;            // truncation == floor for v >= 0
        if (idx > NB - 1) idx = NB - 1;  // right edge -> last bin
        atomicAdd(&sub[idx], 1u);        // ds_add_u32 (no return)
    }
}

__global__ void __launch_bounds__(256)
mi_hist_kernel(const float* __restrict__ x, const float* __restrict__ y,
               unsigned int* __restrict__ gh, int per_batch) {
    // 32 bank-padded sub-histograms: 32*65*4 = 8320 B of the WGP's 320 KB LDS.
    __shared__ unsigned int sh[NSUB * SUBSTRIDE];

    const int tid = threadIdx.x;
    for (int i = tid; i < NSUB * SUBSTRIDE; i += blockDim.x) sh[i] = 0u;
    __syncthreads();

    const int b = blockIdx.y;                // batch index 0..15
    // One sub-histogram per 8 consecutive lanes; stride-65 padding makes
    // equal bins in different subs land on different LDS banks.
    unsigned int* __restrict__ sub = sh + (tid >> 3) * SUBSTRIDE;

    const v4f* __restrict__ x4 =
        (const v4f*)(x + (size_t)b * (size_t)per_batch);
    const v4f* __restrict__ y4 =
        (const v4f*)(y + (size_t)b * (size_t)per_batch);
    const int n4 = per_batch >> 2;

    #pragma unroll 2
    for (int i = blockIdx.x * blockDim.x + tid; i < n4;
         i += gridDim.x * blockDim.x) {
        // Single-pass streaming data: non-temporal b128 loads (TH=NT).
        v4f xv = __builtin_nontemporal_load(&x4[i]);
        v4f yv = __builtin_nontemporal_load(&y4[i]);
        mi_bin_one(xv.x, yv.x, sub);
        mi_bin_one(xv.y, yv.y, sub);
        mi_bin_one(xv.z, yv.z, sub);
        mi_bin_one(xv.w, yv.w, sub);
    }
    __syncthreads();

    // Merge the 32 sub-histograms; one global atomic per bin per block.
    for (int bin = tid; bin < NB; bin += blockDim.x) {
        unsigned int s = 0u;
        #pragma unroll
        for (int c = 0; c < NSUB; ++c) s += sh[c * SUBSTRIDE + bin];
        if (s) atomicAdd(&gh[b * NB + bin], s);   // global_atomic_add_u32
    }
}

__global__ void __launch_bounds__(64)
mi_finalize_kernel(const unsigned int* __restrict__ gh, float* __restrict__ out) {
    __shared__ float red[NB];
    const int t = threadIdx.x;   // 64 threads, one per bin
    float total_mi = 0.0f;

    for (int b = 0; b < NBATCH; ++b) {
        float c = (float)gh[b * NB + t];

        // total = sum(hist)
        red[t] = c;
        __syncthreads();
        if (t == 0) { float s = 0.f; for (int i = 0; i < NB; ++i) s += red[i]; red[0] = s; }
        __syncthreads();
        float total = red[0];
        __syncthreads();

        float jp = c / total + MI_EPS;

        // p = sum(joint_prob)
        red[t] = jp;
        __syncthreads();
        if (t == 0) { float s = 0.f; for (int i = 0; i < NB; ++i) s += red[i]; red[0] = s; }
        __syncthreads();
        float p = red[0];
        __syncthreads();

        // mi_b = sum(jp * log(jp / p^2))
        float term = jp * logf(jp / (p * p));
        red[t] = term;
        __syncthreads();
        if (t == 0) { float s = 0.f; for (int i = 0; i < NB; ++i) s += red[i]; red[0] = s; }
        __syncthreads();
        if (t == 0) total_mi += red[0];
        __syncthreads();
    }

    if (t == 0) out[0] = -(total_mi / (float)NBATCH);
}

extern "C" void kernel_launch(void* const* d_in, const int* in_sizes, int n_in,
                              void* d_out, int out_size, void* d_ws, size_t ws_size,
                              hipStream_t stream) {
    const float* x = (const float*)d_in[0];
    const float* y = (const float*)d_in[1];
    float* out = (float*)d_out;
    unsigned int* gh = (unsigned int*)d_ws;   // 16*64 u32 = 4 KB of scratch

    const int total     = in_sizes[0];            // 16 * 1024 * 1024
    const int per_batch = total / NBATCH;         // 1048576

    // 1) zero the global histogram (ws is not re-poisoned between replays)
    mi_hist_zero_kernel<<<(NBATCH * NB + 255) / 256, 256, 0, stream>>>(gh);

    // 2) bandwidth-bound histogram: 256 blocks/batch x 16 batches x 256 threads
    dim3 grid(256, NBATCH, 1);
    mi_hist_kernel<<<grid, 256, 0, stream>>>(x, y, gh, per_batch);

    // 3) tiny finalize: one 64-thread block reproduces the reference math
    mi_finalize_kernel<<<1, 64, 0, stream>>>(gh, out);
}